// GCNNControl_55920474194385
// MI455X (gfx1250) — compile-verified
//
#include <hip/hip_runtime.h>
#include <hip/hip_bf16.h>
#include <math.h>

#define NNODES 50000
#define DEG    16
#define BATCH  32
#define NDRV   500
#define FEAT   5
#define NPASS  4

typedef float v2f __attribute__((ext_vector_type(2)));
typedef float v8f __attribute__((ext_vector_type(8)));

// ---- per-wave LDS layout (floats) ----
// Feature maps stored [c][w][inst]: addr = c*CS + w*WS + inst, inst innermost
// so the WMMA D scatter (8 consecutive instances per lane) is 2x ds_store_b128.
// WS=40: position stride; 2*WS mod 64 = 16 keeps lo/hi half K-slots of one
// A-load 16 banks apart. CS = W*WS+4 keeps channel bases 16B-aligned.
#define WS        40
#define CS_OF(W)  ((W) * WS + 4)
#define G_SZ      (16 * WS)               // gathered input [16 w][32 inst] = 640
#define HA_SZ     (FEAT * CS_OF(14))      // 2820 (holds W=14,10,6)
#define HB_SZ     (FEAT * CS_OF(12))      // 2420 (holds W=12,8,4)
#define DUMP_F    112                     // per-lane 4-float sinks for pad lanes
#define WAVE_F    (G_SZ + HA_SZ + HB_SZ + DUMP_F)   // 5992 floats / wave

// single-instruction ReLU (avoids the canonicalize+max pair maxnum lowers to)
__device__ __forceinline__ float relu1(float x) {
  asm("v_max_num_f32 %0, 0, %0" : "+v"(x));
  return x;
}

// One conv layer (kernel width 3, VALID) for 32 instances, via
// V_WMMA_F32_16X16X4_F32:  D[inst, c_out] += A[inst, k] * B[k, c_out],
// K = CIN*3 (zero-padded in B to 4-chunks), one WMMA chain per output position.
// A operands double-buffered, both 16-instance tiles interleaved. Stores are
// unconditional (padding lanes write private dump slots selected per position
// so the w*WS immediate never touches the dump) -> deterministic DS counts,
// partial s_wait_dscnt, no EXEC churn.
template<int WIN, int CIN>
__device__ __forceinline__ void conv_layer(const float* hin, int csIn,
                                           float* hout, int csOut,
                                           float* dump,
                                           const float* wgt,   // [c_out][CIN][3]
                                           const float* bias,  // [c_out]
                                           int lane) {
  constexpr int NK   = CIN * 3;
  constexpr int NCH  = (NK + 3) / 4;
  constexpr int WOUT = WIN - 2;
  const int  m  = lane & 15;       // A row (instance-in-tile) and D column (c_out)
  const bool lo = lane < 16;

  // B operand (weights) + A address offsets, per K-chunk.
  // f32 A 16x4 layout: vgpr0 holds K=kc*4+{0|2}, vgpr1 holds K=kc*4+{1|3} (lo|hi half)
  v2f Bv[NCH];
  int Aoff[NCH][2];
#pragma unroll
  for (int kc = 0; kc < NCH; ++kc) {
#pragma unroll
    for (int v = 0; v < 2; ++v) {
      int k  = kc * 4 + v + (lo ? 0 : 2);
      int kk = (k < NK) ? k : (NK - 1);      // clamp address (value masked by B=0)
      int ci = kk / 3, t = kk - ci * 3;
      Aoff[kc][v] = ci * csIn + t * WS + m;
      Bv[kc][v] = (m < FEAT && k < NK) ? wgt[(m * CIN + ci) * 3 + t] : 0.0f;
    }
  }
  float bval = (m < FEAT) ? bias[m] : 0.0f;

  // store base: lane m = c_out, D vgpr r -> instance r (+8 hi half) (+16 tile 1)
  float* so  = hout + m * csOut + (lo ? 0 : 8);
  float* dmp = dump + ((m >= FEAT) ? (m - FEAT + (lo ? 0 : 11)) * 4 : 0);

  v2f a0[NCH], a1[NCH];
#pragma unroll
  for (int kc = 0; kc < NCH; ++kc) {
    a0[kc][0] = hin[Aoff[kc][0]];      a0[kc][1] = hin[Aoff[kc][1]];
    a1[kc][0] = hin[Aoff[kc][0] + 16]; a1[kc][1] = hin[Aoff[kc][1] + 16];
  }

#pragma unroll
  for (int w = 0; w < WOUT; ++w) {
    v2f n0[NCH], n1[NCH];
    if (w + 1 < WOUT) {                      // prefetch next position's A operands
#pragma unroll
      for (int kc = 0; kc < NCH; ++kc) {
        int o0 = Aoff[kc][0] + (w + 1) * WS;
        int o1 = Aoff[kc][1] + (w + 1) * WS;
        n0[kc][0] = hin[o0];      n0[kc][1] = hin[o1];
        n1[kc][0] = hin[o0 + 16]; n1[kc][1] = hin[o1 + 16];
      }
    }
    v8f acc0, acc1;
#pragma unroll
    for (int e = 0; e < 8; ++e) { acc0[e] = bval; acc1[e] = bval; }  // bias in C
#pragma unroll
    for (int kc = 0; kc < NCH; ++kc)
      acc0 = __builtin_amdgcn_wmma_f32_16x16x4_f32(
                 false, a0[kc], false, Bv[kc], (short)0, acc0, false, false);
#pragma unroll
    for (int kc = 0; kc < NCH; ++kc)
      acc1 = __builtin_amdgcn_wmma_f32_16x16x4_f32(
                 false, a1[kc], false, Bv[kc], (short)0, acc1, false, false);
#pragma unroll
    for (int e = 0; e < 8; ++e) {            // ReLU, 1 v_max_num each
      acc0[e] = relu1(acc0[e]);
      acc1[e] = relu1(acc1[e]);
    }
    // branchless store: pad lanes -> private dump slot (no w*WS on dump side)
    float* sp = (m < FEAT) ? (so + w * WS) : dmp;
#pragma unroll
    for (int r = 0; r < 8; ++r) sp[r] = acc0[r];        // 2x ds_store_b128
#pragma unroll
    for (int r = 0; r < 8; ++r) sp[16 + r] = acc1[r];   // 2x ds_store_b128
    if (w + 1 < WOUT) {
#pragma unroll
      for (int kc = 0; kc < NCH; ++kc) { a0[kc] = n0[kc]; a1[kc] = n1[kc]; }
    }
  }
}

__global__ __launch_bounds__(64)
void k_pass(const float* __restrict__ zin,   // [N][B]
            float* __restrict__ zout,        // [N][B]
            const int* __restrict__ nidx,    // [N][16]
            const float* __restrict__ w_first, const float* __restrict__ b_first,
            const float* __restrict__ w_mid,   const float* __restrict__ b_mid,
            const float* __restrict__ w_last,  const float* __restrict__ b_last) {
  __shared__ float lds_pool[2 * WAVE_F] __attribute__((aligned(16)));
  const int wave = threadIdx.x >> 5;
  const int lane = threadIdx.x & 31;
  const int node = blockIdx.x * 2 + wave;
  if (node >= NNODES) return;

  float* G    = lds_pool + wave * WAVE_F;    // [w][inst], stride WS
  float* HA   = G + G_SZ;                    // [c][w][inst], CS of current width
  float* HB   = HA + HA_SZ;
  float* DUMP = HB + HB_SZ;

  // Gather: scalar neighbor indices, fully coalesced 128B lines from zT[N][B];
  // G store: 32 consecutive lanes -> conflict-free
  const int* row = nidx + node * DEG;
#pragma unroll
  for (int j = 0; j < DEG; ++j) {
    int nb = row[j];                          // wave-uniform -> SGPR
    G[j * WS + lane] = zin[nb * BATCH + lane];
  }
  // same-wave LDS producer/consumer: DS ops are in-order per wave

  conv_layer<16, 1>(G,  0,          HA, CS_OF(14), DUMP, w_first,     b_first,    lane);
  conv_layer<14, FEAT>(HA, CS_OF(14), HB, CS_OF(12), DUMP, w_mid +   0, b_mid +  0, lane);
  conv_layer<12, FEAT>(HB, CS_OF(12), HA, CS_OF(10), DUMP, w_mid +  75, b_mid +  5, lane);
  conv_layer<10, FEAT>(HA, CS_OF(10), HB, CS_OF(8),  DUMP, w_mid + 150, b_mid + 10, lane);
  conv_layer< 8, FEAT>(HB, CS_OF(8),  HA, CS_OF(6),  DUMP, w_mid + 225, b_mid + 15, lane);
  conv_layer< 6, FEAT>(HA, CS_OF(6),  HB, CS_OF(4),  DUMP, w_mid + 300, b_mid + 20, lane);

  // Last conv (5->1, k3, W=4->2) + ReLU + avgpool(2) in VALU (tiny)
  const float* hf = HB + lane;                // inst = lane; conflict-free reads
  float bl = b_last[0];
  float o0 = bl, o1 = bl;
#pragma unroll
  for (int c = 0; c < FEAT; ++c) {
#pragma unroll
    for (int t = 0; t < 3; ++t) {
      float wv = w_last[c * 3 + t];
      o0 += wv * hf[c * CS_OF(4) + t * WS];
      o1 += wv * hf[c * CS_OF(4) + (t + 1) * WS];
    }
  }
  o0 = relu1(o0);
  o1 = relu1(o1);
  zout[node * BATCH + lane] = 0.5f * (o0 + o1);
}

__global__ void k_transpose(const float* __restrict__ x, float* __restrict__ zT) {
  int i = blockIdx.x * blockDim.x + threadIdx.x;   // i = n*B + b
  if (i < NNODES * BATCH) {
    int n = i >> 5, b = i & 31;
    zT[i] = x[b * NNODES + n];
  }
}

__global__ void k_fill0(float* __restrict__ out, int n) {
  int i = blockIdx.x * blockDim.x + threadIdx.x;
  if (i < n) out[i] = 0.0f;
}

// softmax over u[b,:] where u = z at driver nodes, 0 elsewhere; masked output
__global__ void k_softmax(const float* __restrict__ zT,   // [N][B]
                          const int* __restrict__ drivers,
                          float* __restrict__ out) {      // [B][N]
  const int b = blockIdx.x, tid = threadIdx.x;
  __shared__ float red[256];

  float mloc = 0.0f;                                  // baseline: N-500 zeros
  for (int d = tid; d < NDRV; d += 256)
    mloc = fmaxf(mloc, zT[drivers[d] * BATCH + b]);
  red[tid] = mloc; __syncthreads();
  for (int s = 128; s > 0; s >>= 1) {
    if (tid < s) red[tid] = fmaxf(red[tid], red[tid + s]);
    __syncthreads();
  }
  float m = red[0];
  __syncthreads();

  float sloc = 0.0f;
  for (int d = tid; d < NDRV; d += 256)
    sloc += expf(zT[drivers[d] * BATCH + b] - m);
  red[tid] = sloc; __syncthreads();
  for (int s = 128; s > 0; s >>= 1) {
    if (tid < s) red[tid] += red[tid + s];
    __syncthreads();
  }
  float den = red[0] + (float)(NNODES - NDRV) * expf(-m);

  for (int d = tid; d < NDRV; d += 256) {
    int n = drivers[d];
    out[b * NNODES + n] = expf(zT[n * BATCH + b] - m) / den;
  }
}

extern "C" void kernel_launch(void* const* d_in, const int* in_sizes, int n_in,
                              void* d_out, int out_size, void* d_ws, size_t ws_size,
                              hipStream_t stream) {
  (void)in_sizes; (void)n_in; (void)out_size; (void)ws_size;
  const float* x       = (const float*)d_in[0];
  const int*   nidx    = (const int*)  d_in[1];
  const int*   drivers = (const int*)  d_in[2];
  // d_in[3] = driver_matrix: one-hot, collapsed analytically (unused)
  const float* w_first = (const float*)d_in[4];
  const float* b_first = (const float*)d_in[5];
  const float* w_mid   = (const float*)d_in[6];
  const float* b_mid   = (const float*)d_in[7];
  const float* w_last  = (const float*)d_in[8];
  const float* b_last  = (const float*)d_in[9];
  float* out = (float*)d_out;

  float* zA = (float*)d_ws;                 // zT ping [N][B]
  float* zB = zA + NNODES * BATCH;          // zT pong [N][B]

  k_transpose<<<(NNODES * BATCH + 255) / 256, 256, 0, stream>>>(x, zA);

  for (int p = 0; p < NPASS; ++p) {
    const float* zi = (p & 1) ? zB : zA;
    float*       zo = (p & 1) ? zA : zB;
    k_pass<<<NNODES / 2, 64, 0, stream>>>(zi, zo, nidx,
                                          w_first, b_first, w_mid, b_mid,
                                          w_last, b_last);
  }
  // after pass 3 (odd), final state is in zA

  k_fill0<<<(NNODES * BATCH + 255) / 256, 256, 0, stream>>>(out, NNODES * BATCH);
  k_softmax<<<BATCH, 256, 0, stream>>>(zA, drivers, out);
}